// GraphRecommendationModel_86758339379706
// MI455X (gfx1250) — compile-verified
//
#include <hip/hip_runtime.h>

#define B_  256
#define L_  50
#define N_  100
#define D_  64

typedef float v2f __attribute__((ext_vector_type(2)));
typedef float v8f __attribute__((ext_vector_type(8)));
typedef int   i32x4 __attribute__((vector_size(16)));   // matches builtin param

// ---------------------------------------------------------------------------
// CDNA5 async copy: 16 bytes global -> LDS, tracked with ASYNCcnt.
// Builtin takes typed pointers: (v4i AS1* src, v4i AS3* dst, imm off, imm cpol)
// ---------------------------------------------------------------------------
__device__ __forceinline__ void async_copy_b128(const float* g, float* l) {
#if __has_builtin(__builtin_amdgcn_global_load_async_to_lds_b128)
  __builtin_amdgcn_global_load_async_to_lds_b128(
      (__attribute__((address_space(1))) i32x4*)(void*)g,
      (__attribute__((address_space(3))) i32x4*)(void*)l,
      /*offset=*/0, /*cpol=*/0);
#else
  *(float4*)l = *(const float4*)g;   // fallback: sync copy through VGPRs
#endif
}

__device__ __forceinline__ void wait_async_zero() {
#if __has_builtin(__builtin_amdgcn_s_wait_asynccnt)
  __builtin_amdgcn_s_wait_asynccnt(0);
#else
  asm volatile("s_wait_asynccnt 0" ::: "memory");
#endif
}

// ---------------------------------------------------------------------------
// V_WMMA_F32_16X16X4_F32 wrapper (8-arg VOP3P form).
// ---------------------------------------------------------------------------
__device__ __forceinline__ v8f wmma_16x16x4_f32(v2f a, v2f b, v8f c) {
  return __builtin_amdgcn_wmma_f32_16x16x4_f32(
      false, a, false, b, (short)0, c, false, false);
}

// ---------------------------------------------------------------------------
// Kernel A: one block per (b,l) pair. Stages the 100x64 neighbor tile into
// LDS via async-to-LDS, computes GAT logits, softmax over N, and the
// attention-weighted aggregation. Writes agg_nb row (64 floats) to ws.
// This kernel is the entire bandwidth cost (327.7 MB streamed once).
// ---------------------------------------------------------------------------
__launch_bounds__(128)
__global__ void gr_attn_kernel(const int*   __restrict__ history,       // (B*L)
                               const float* __restrict__ neighbor_emb,  // (B,L,N,D)
                               const float* __restrict__ item_table,
                               const float* __restrict__ neighbor_gat_w,// (128)
                               const float* __restrict__ neighbor_gat_b,// (1)
                               float*       __restrict__ agg_nb_out)    // (B*L,64)
{
  __shared__ float nb[N_ * D_];   // 25600 B neighbor tile
  __shared__ float nr_s[D_];
  __shared__ float sarr[N_];
  __shared__ float red[D_];
  __shared__ float sleft_s;

  const int pair = blockIdx.x;        // b*L + l
  const int tid  = threadIdx.x;       // 0..127

  // ---- issue async global->LDS copy of the tile (1600 x b128) ----
  const float* gsrc = neighbor_emb + (size_t)pair * (N_ * D_);
  for (int i = tid; i < (N_ * D_) / 4; i += 128)
    async_copy_b128(gsrc + i * 4, nb + i * 4);

  // ---- overlap: agg_real scalar + s_left pieces (real-GAT softmax ~= c) ----
  float csum = 0.f;
  #pragma unroll
  for (int i = 0; i < L_; ++i) csum += (1.0f / (float)L_);

  if (tid < D_) {
    const int hidx = history[pair];
    const float hv = item_table[(size_t)hidx * D_ + tid] * csum; // agg_real[d]
    nr_s[tid] = neighbor_gat_w[D_ + tid];                        // nr
    red[tid]  = hv * neighbor_gat_w[tid];                        // h*c . nl
  }
  __syncthreads();
  if (tid == 0) {
    float s = 0.f;
    #pragma unroll
    for (int i = 0; i < D_; ++i) s += red[i];
    sleft_s = s;
  }

  wait_async_zero();     // drain ASYNCcnt for this wave's tile chunks
  __syncthreads();       // tile + sleft_s visible to all waves

  // ---- logits: s_nb[n] = relu(s_left + neigh[n].nr + bias) ----
  const float bias = neighbor_gat_b[0];
  if (tid < N_) {
    const float sl = sleft_s;
    const int   base = tid * D_;
    float acc = 0.f;
    #pragma unroll 8
    for (int j = 0; j < D_; ++j) {
      const int idx = (tid + j) & (D_ - 1);   // rotation: bank-conflict free
      acc += nb[base + idx] * nr_s[idx];
    }
    sarr[tid] = fmaxf(sl + acc + bias, 0.f);
  }
  __syncthreads();

  // ---- softmax over N=100 (redundant per-thread stats; broadcast reads) ----
  float mx = -3.0e38f;
  for (int n = 0; n < N_; ++n) mx = fmaxf(mx, sarr[n]);
  __syncthreads();
  if (tid < N_) sarr[tid] = __expf(sarr[tid] - mx);
  __syncthreads();
  float ssum = 0.f;
  for (int n = 0; n < N_; ++n) ssum += sarr[n];

  // ---- aggregation: agg_nb[d] = (1/ssum) * sum_n e[n]*neigh[n,d] ----
  if (tid < D_) {
    float acc = 0.f;
    #pragma unroll 4
    for (int n = 0; n < N_; ++n) acc += sarr[n] * nb[n * D_ + tid];
    agg_nb_out[(size_t)pair * D_ + tid] = acc / ssum;
  }
}

// ---------------------------------------------------------------------------
// Kernel B: one block per batch row b (64 threads = thread per feature d).
// Mean over L of agg_nb -> abar; mean over L of agg_real; builds the
// "combined" row [u | realmean(+mh later) | it] in ws.
// ---------------------------------------------------------------------------
__launch_bounds__(64)
__global__ void gr_reduce_kernel(const int*   __restrict__ user_ids,
                                 const int*   __restrict__ item_ids,
                                 const int*   __restrict__ history,
                                 const float* __restrict__ user_table,
                                 const float* __restrict__ item_table,
                                 const float* __restrict__ agg_nb,   // (B*L,64)
                                 float*       __restrict__ abar,     // (B,64)
                                 float*       __restrict__ comb)     // (B,192)
{
  const int b = blockIdx.x;
  const int d = threadIdx.x;

  float csum = 0.f;
  #pragma unroll
  for (int i = 0; i < L_; ++i) csum += (1.0f / (float)L_);

  float accA = 0.f, accR = 0.f;
  for (int l = 0; l < L_; ++l) {
    accA += agg_nb[((size_t)(b * L_ + l)) * D_ + d];
    const int hidx = history[b * L_ + l];
    accR += item_table[(size_t)hidx * D_ + d];
  }
  abar[b * D_ + d] = accA * (1.0f / (float)L_);

  comb[b * 192 + d]        = user_table[(size_t)user_ids[b] * D_ + d];
  comb[b * 192 + 64 + d]   = accR * (csum / (float)L_);   // mean(agg_real)
  comb[b * 192 + 128 + d]  = item_table[(size_t)item_ids[b] * D_ + d];
}

// ---------------------------------------------------------------------------
// Kernel C: WMMA tail. 16 blocks x 1 wave; block handles 16 batch rows.
//   aggregated += abar @ mh_w^T + mh_b        (16x64 @ 64x64)
//   x1 = relu(combined @ fc1_w^T + fc1_b)     (16x192 @ 192x64)
//   x2 = relu(x1 @ fc2_w^T + fc2_b)           (16x64 @ 64x64)
//   out = sigmoid(x2 @ out_w^T + out_b)*4+1
// Fragment layouts per CDNA5 ISA: A 16x4 (row=lane&15, K pair by lane>=16),
// B 4x16 (col=lane&15), C/D 8 VGPRs (row = i + 8*(lane>=16), col = lane&15).
// ---------------------------------------------------------------------------
__launch_bounds__(32)
__global__ void gr_head_kernel(const float* __restrict__ abar,   // (B,64)
                               const float* __restrict__ comb,   // (B,192)
                               const float* __restrict__ mh_w,   // (64,64) flat
                               const float* __restrict__ mh_b,   // (64)
                               const float* __restrict__ fc1_w,  // (64,192)
                               const float* __restrict__ fc1_b,  // (64)
                               const float* __restrict__ fc2_w,  // (64,64)
                               const float* __restrict__ fc2_b,  // (64)
                               const float* __restrict__ out_w,  // (64)
                               const float* __restrict__ out_b,  // (1)
                               float*       __restrict__ out)    // (B)
{
  __shared__ float cmb[16][196];   // padded: 196 % 64 = 4 -> no bank clash
  __shared__ float x1[16][68];
  __shared__ float x2[16][68];

  const int lane = threadIdx.x;          // 0..31, one full wave
  const int m0   = blockIdx.x * 16;
  const int rlo  = lane & 15;
  const int khi  = (lane >> 4) * 2;      // K sub-offset 0 or 2
  const int rofs = (lane >> 4) * 8;      // C/D row offset 0 or 8

  // load combined tile (aggregated cols currently hold mean(agg_real))
  for (int i = lane; i < 16 * 192; i += 32) {
    const int r = i / 192, c = i % 192;
    cmb[r][c] = comb[(size_t)(m0 + r) * 192 + c];
  }
  __syncthreads();

  // ---- mh: aggregated += abar @ mh_w^T + mh_b ----
  for (int nc = 0; nc < 64; nc += 16) {
    v8f acc = {};
    const int col = nc + rlo;
    for (int k = 0; k < 64; k += 4) {
      const int kb = k + khi;
      v2f a, b;
      a.x = abar[(size_t)(m0 + rlo) * 64 + kb];
      a.y = abar[(size_t)(m0 + rlo) * 64 + kb + 1];
      b.x = mh_w[col * 64 + kb];
      b.y = mh_w[col * 64 + kb + 1];
      acc = wmma_16x16x4_f32(a, b, acc);
    }
    const float mb = mh_b[col];
    #pragma unroll
    for (int i = 0; i < 8; ++i)
      cmb[i + rofs][64 + col] += acc[i] + mb;
  }
  __syncthreads();

  // ---- fc1: x1 = relu(combined @ fc1_w^T + b1), K=192 ----
  for (int nc = 0; nc < 64; nc += 16) {
    v8f acc = {};
    const int col = nc + rlo;
    for (int k = 0; k < 192; k += 4) {
      const int kb = k + khi;
      v2f a, b;
      a.x = cmb[rlo][kb];
      a.y = cmb[rlo][kb + 1];
      b.x = fc1_w[col * 192 + kb];
      b.y = fc1_w[col * 192 + kb + 1];
      acc = wmma_16x16x4_f32(a, b, acc);
    }
    const float bb = fc1_b[col];
    #pragma unroll
    for (int i = 0; i < 8; ++i)
      x1[i + rofs][col] = fmaxf(acc[i] + bb, 0.f);
  }
  __syncthreads();

  // ---- fc2: x2 = relu(x1 @ fc2_w^T + b2), K=64 ----
  for (int nc = 0; nc < 64; nc += 16) {
    v8f acc = {};
    const int col = nc + rlo;
    for (int k = 0; k < 64; k += 4) {
      const int kb = k + khi;
      v2f a, b;
      a.x = x1[rlo][kb];
      a.y = x1[rlo][kb + 1];
      b.x = fc2_w[col * 64 + kb];
      b.y = fc2_w[col * 64 + kb + 1];
      acc = wmma_16x16x4_f32(a, b, acc);
    }
    const float bb = fc2_b[col];
    #pragma unroll
    for (int i = 0; i < 8; ++i)
      x2[i + rofs][col] = fmaxf(acc[i] + bb, 0.f);
  }
  __syncthreads();

  // ---- output head: sigmoid(x2 . out_w + out_b) * 4 + 1 ----
  if (lane < 16) {
    float acc = 0.f;
    #pragma unroll 8
    for (int d = 0; d < 64; ++d) acc += x2[lane][d] * out_w[d];
    const float v = acc + out_b[0];
    out[m0 + lane] = 4.0f / (1.0f + __expf(-v)) + 1.0f;
  }
}

// ---------------------------------------------------------------------------
extern "C" void kernel_launch(void* const* d_in, const int* in_sizes, int n_in,
                              void* d_out, int out_size, void* d_ws, size_t ws_size,
                              hipStream_t stream) {
  (void)in_sizes; (void)n_in; (void)out_size; (void)ws_size;

  const int*   user_ids       = (const int*)  d_in[0];
  const int*   item_ids       = (const int*)  d_in[1];
  const int*   history        = (const int*)  d_in[2];
  const float* neighbor_emb   = (const float*)d_in[3];
  const float* user_table     = (const float*)d_in[4];
  const float* item_table     = (const float*)d_in[5];
  // d_in[6]/d_in[7] (real_gat_w/b) cancel analytically: softmax rows sum to 1.
  const float* neighbor_gat_w = (const float*)d_in[8];
  const float* neighbor_gat_b = (const float*)d_in[9];
  const float* mh_w           = (const float*)d_in[10];
  const float* mh_b           = (const float*)d_in[11];
  const float* fc1_w          = (const float*)d_in[12];
  const float* fc1_b          = (const float*)d_in[13];
  const float* fc2_w          = (const float*)d_in[14];
  const float* fc2_b          = (const float*)d_in[15];
  const float* out_w          = (const float*)d_in[16];
  const float* out_b          = (const float*)d_in[17];
  float* out = (float*)d_out;

  // workspace layout (floats): agg_nb (B*L*64) | abar (B*64) | comb (B*192)
  float* ws     = (float*)d_ws;
  float* agg_nb = ws;                                   // 819200 f
  float* abar   = ws + (size_t)B_ * L_ * D_;            //  16384 f
  float* comb   = abar + (size_t)B_ * D_;               //  49152 f

  gr_attn_kernel<<<B_ * L_, 128, 0, stream>>>(
      history, neighbor_emb, item_table, neighbor_gat_w, neighbor_gat_b, agg_nb);

  gr_reduce_kernel<<<B_, 64, 0, stream>>>(
      user_ids, item_ids, history, user_table, item_table, agg_nb, abar, comb);

  gr_head_kernel<<<B_ / 16, 32, 0, stream>>>(
      abar, comb, mh_w, mh_b, fc1_w, fc1_b, fc2_w, fc2_b, out_w, out_b, out);
}